// BasicModel_24429773979973
// MI455X (gfx1250) — compile-verified
//
#include <hip/hip_runtime.h>

#define BB 8192
#define NN 512
#define LL 5

__global__ __launch_bounds__(256) void qsim_rx_rz_kernel(
    const float* __restrict__ x,      // (B, N)
    const float* __restrict__ fRx,    // (L, N)
    const float* __restrict__ fRz,    // (L, N)
    float* __restrict__ state_out,    // B*N * 4 floats  (s0.re, s0.im, s1.re, s1.im)
    float* __restrict__ O_out)        // B*N * 2 floats  (O.re, 0)
{
    // Per-(layer,qubit) constant tables in LDS (40 KB of the 320 KB WGP pool).
    __shared__ float CR [LL * NN];   // cos(rx_j/2)   (staging: raw fRx)
    __shared__ float SR [LL * NN];   // sin(rx_j/2)
    __shared__ float PZR[LL * NN];   // cos(rz_j/2)   (staging: raw fRz)
    __shared__ float PZI[LL * NN];   // -sin(rz_j/2)

    // ---- Stage theta arrays into LDS via CDNA5 async global->LDS DMA ----
    // Tracked by ASYNCcnt; each lane DMAs 4 bytes into its own LDS slot.
    for (int i = threadIdx.x; i < LL * NN; i += blockDim.x) {
        unsigned lds_rx = (unsigned)(unsigned long long)&CR[i];   // low 32b of flat = LDS offset
        unsigned lds_rz = (unsigned)(unsigned long long)&PZR[i];
        unsigned goff   = (unsigned)(i * 4);
        asm volatile("global_load_async_to_lds_b32 %0, %1, %2"
                     :: "v"(lds_rx), "v"(goff), "s"(fRx) : "memory");
        asm volatile("global_load_async_to_lds_b32 %0, %1, %2"
                     :: "v"(lds_rz), "v"(goff), "s"(fRz) : "memory");
    }
    asm volatile("s_wait_asynccnt 0x0" ::: "memory");

    // In-place convert raw thetas -> (cos, sin) tables. Each wave converts the
    // entries it DMA'd itself, so no cross-wave hazard until the barrier.
    for (int i = threadIdx.x; i < LL * NN; i += blockDim.x) {
        float trx = CR[i];
        float trz = PZR[i];
        float s, c;
        __sincosf(0.5f * trx, &s, &c);
        CR[i]  = c;
        SR[i]  = s;
        __sincosf(0.5f * trz, &s, &c);
        PZR[i] = c;
        PZI[i] = -s;   // ph = exp(-i t/2) = cos - i sin
    }
    __syncthreads();

    // ---- Streaming main loop: 1 sincos + 5 fused layers per element ----
    const int total  = BB * NN;
    const int stride = (int)(gridDim.x * blockDim.x);
    for (int idx = (int)(blockIdx.x * blockDim.x + threadIdx.x); idx < total; idx += stride) {
        if (idx + stride < total) __builtin_prefetch(x + idx + stride, 0, 0);

        const float xv = x[idx];
        const int   n  = idx & (NN - 1);

        float cx, sx;
        __sincosf(0.5f * xv, &sx, &cx);

        // state = |0>
        float s0r = 1.0f, s0i = 0.0f, s1r = 0.0f, s1i = 0.0f;

#pragma unroll
        for (int j = 0; j < LL; ++j) {
            const int k  = j * NN + n;
            const float cr = CR[k], sr = SR[k];
            // angle addition: c = cos((rx_j + x)/2), s = sin((rx_j + x)/2)
            const float c = __builtin_fmaf(-sr, sx, cr * cx);
            const float s = __builtin_fmaf( sr, cx, cr * sx);
            // Rx: s0' = c*s0 - i*s*s1 ; s1' = -i*s*s0 + c*s1
            const float t0r = __builtin_fmaf(c, s0r,  s * s1i);
            const float t0i = __builtin_fmaf(c, s0i, -s * s1r);
            const float t1r = __builtin_fmaf(c, s1r,  s * s0i);
            const float t1i = __builtin_fmaf(c, s1i, -s * s0r);
            // Rz: s0 *= (pr + i*pi), s1 *= conj(pr + i*pi)
            const float pr = PZR[k], pi = PZI[k];
            s0r = __builtin_fmaf(t0r, pr, -t0i * pi);
            s0i = __builtin_fmaf(t0r, pi,  t0i * pr);
            s1r = __builtin_fmaf(t1r, pr,  t1i * pi);
            s1i = __builtin_fmaf(t1i, pr, -t1r * pi);
        }

        // state: (B, N, 1, 2, 1) complex64 -> 4 consecutive floats, B128 store
        float4 st;
        st.x = s0r; st.y = s0i; st.z = s1r; st.w = s1i;
        reinterpret_cast<float4*>(state_out)[idx] = st;

        // O = |s0|^2 - |s1|^2 (complex64 with zero imag), B64 store
        float2 o;
        o.x = __builtin_fmaf(s0r, s0r, s0i * s0i) - __builtin_fmaf(s1r, s1r, s1i * s1i);
        o.y = 0.0f;
        reinterpret_cast<float2*>(O_out)[idx] = o;
    }
}

extern "C" void kernel_launch(void* const* d_in, const int* in_sizes, int n_in,
                              void* d_out, int out_size, void* d_ws, size_t ws_size,
                              hipStream_t stream) {
    (void)in_sizes; (void)n_in; (void)out_size; (void)d_ws; (void)ws_size;
    const float* x   = (const float*)d_in[0];
    const float* fRx = (const float*)d_in[1];
    const float* fRz = (const float*)d_in[2];

    float* out   = (float*)d_out;
    float* state = out;                              // B*N*4 floats
    float* O     = out + (size_t)BB * NN * 4;        // B*N*2 floats

    dim3 block(256);    // 8 wave32s per block
    dim3 grid(2048);    // grid-stride: ~8 elements/thread, amortizes LDS table build
    hipLaunchKernelGGL(qsim_rx_rz_kernel, grid, block, 0, stream,
                       x, fRx, fRz, state, O);
}